// PureGINConv_66340064854628
// MI455X (gfx1250) — compile-verified
//
#include <hip/hip_runtime.h>
#include <hip/hip_bf16.h>

#define N_NODES_C 100000
#define D_C 64
#define H_C 128
#define WAVES_PB 4
#define TILES_PW 8   // tiles per wave; block covers WAVES_PB*TILES_PW = 32 tiles

typedef __attribute__((ext_vector_type(2))) float v2f;
typedef __attribute__((ext_vector_type(8))) float v8f;

// ---------------------------------------------------------------------------
// Kernel 1: agg = (1 + eps) * x  (eps = 0 -> copy), vectorized float4
// ---------------------------------------------------------------------------
__global__ __launch_bounds__(256) void init_agg_kernel(const float4* __restrict__ x4,
                                                       float4* __restrict__ agg4,
                                                       int n4) {
    int i = blockIdx.x * blockDim.x + threadIdx.x;
    if (i < n4) agg4[i] = x4[i];
}

// ---------------------------------------------------------------------------
// Kernel 2: scatter-add  agg[dst] += x[src]  — one wave32 per edge,
// each lane owns 2 of the 64 feature columns (coalesced 256B/edge).
// Relaxed agent-scope float fadd -> native global_atomic_add_f32 (L2 atomic).
// ---------------------------------------------------------------------------
__global__ __launch_bounds__(128) void scatter_add_kernel(const float* __restrict__ x,
                                                          const long long* __restrict__ ei,
                                                          float* __restrict__ agg,
                                                          int n_edges) {
    const int wave = blockIdx.x * (blockDim.x >> 5) + (threadIdx.x >> 5);
    const int lane = threadIdx.x & 31;
    if (wave >= n_edges) return;
    const int src = (int)ei[wave];
    const int dst = (int)ei[(size_t)n_edges + wave];
    const float2 v = *(const float2*)(x + (size_t)src * D_C + lane * 2);
    float* p = agg + (size_t)dst * D_C + lane * 2;
    __hip_atomic_fetch_add(p + 0, v.x, __ATOMIC_RELAXED, __HIP_MEMORY_SCOPE_AGENT);
    __hip_atomic_fetch_add(p + 1, v.y, __ATOMIC_RELAXED, __HIP_MEMORY_SCOPE_AGENT);
}

// ---------------------------------------------------------------------------
// Kernel 3: fused GIN MLP with f32 WMMA.
// Weights staged once per block into LDS in "k-pair" layout via async
// global->LDS copies (GLOBAL_LOAD_ASYNC_TO_LDS_B32, ASYNCcnt), so each
// B fragment is a single ds_load_b64. One wave per 16-row tile, 8 tiles/wave.
// ---------------------------------------------------------------------------
__global__ __launch_bounds__(128) void gin_mlp_kernel(const float* __restrict__ agg,
                                                      const float* __restrict__ W1,
                                                      const float* __restrict__ b1,
                                                      const float* __restrict__ W2,
                                                      const float* __restrict__ b2,
                                                      float* __restrict__ out,
                                                      int n_tiles) {
    // W1L[(k/2)*H + c] = {W1[k][c], W1[k+1][c]}   (32*128 float2 = 32KB)
    // W2L[(k/2)*D + c] = {W2[k][c], W2[k+1][c]}   (64*64  float2 = 32KB)
    __shared__ float2 W1L[32 * H_C];
    __shared__ float2 W2L[64 * D_C];
    __shared__ float  hb[WAVES_PB][16 * H_C];      // 4 x 8KB h slabs

    const int tid = threadIdx.x;

    // ---- Stage weights into LDS with async global->LDS copies (transposing
    //      into the k-pair layout via per-lane LDS addresses). ----
    {
        const unsigned w1base = (unsigned)(uintptr_t)(&W1L[0]);
        for (int i = tid; i < D_C * H_C; i += 128) {
            const int k = i >> 7, c = i & (H_C - 1);
            const unsigned dst = w1base + (unsigned)(((((k >> 1) * H_C + c) << 1) + (k & 1)) * 4);
            const float* g = W1 + i;
            asm volatile("global_load_async_to_lds_b32 %0, %1, off"
                         :: "v"(dst), "v"(g) : "memory");
        }
        const unsigned w2base = (unsigned)(uintptr_t)(&W2L[0]);
        for (int i = tid; i < H_C * D_C; i += 128) {
            const int k = i >> 6, c = i & (D_C - 1);
            const unsigned dst = w2base + (unsigned)(((((k >> 1) * D_C + c) << 1) + (k & 1)) * 4);
            const float* g = W2 + i;
            asm volatile("global_load_async_to_lds_b32 %0, %1, off"
                         :: "v"(dst), "v"(g) : "memory");
        }
        asm volatile("s_wait_asynccnt 0x0" ::: "memory");
    }
    __syncthreads();

    const int wave = tid >> 5;
    const int lane = tid & 31;
    const int row  = lane & 15;     // M (A) / N (B,C,D) index within half-wave
    const int hi   = lane >> 4;     // 0: lanes 0-15, 1: lanes 16-31

    float*  hbuf  = &hb[wave][0];
    float2* hbuf2 = (float2*)hbuf;

    // Bias fragments (column-only) hoisted out of the tile loop.
    float bv1[8], bv2[4];
#pragma unroll
    for (int n = 0; n < 8; ++n) bv1[n] = b1[n * 16 + row];
#pragma unroll
    for (int n = 0; n < 4; ++n) bv2[n] = b2[n * 16 + row];

    const int tile0 = blockIdx.x * (WAVES_PB * TILES_PW) + wave * TILES_PW;

    for (int t = 0; t < TILES_PW; ++t) {
        const int tile = tile0 + t;
        if (tile >= n_tiles) break;            // wave-uniform: EXEC stays all-1s
        const int m0 = tile * 16;
        const float* A = agg + (size_t)m0 * D_C;

        // ---- Layer 1: h(16x128) = relu(A(16x64) @ W1 + b1) ----
        v8f acc[8];
#pragma unroll
        for (int n = 0; n < 8; ++n) {
            const float bv = bv1[n];
            acc[n] = (v8f){bv, bv, bv, bv, bv, bv, bv, bv};
        }
#pragma unroll
        for (int k = 0; k < D_C; k += 4) {
            const float2 af = *(const float2*)(A + row * D_C + k + 2 * hi);
            v2f a; a.x = af.x; a.y = af.y;
            const int p = (k >> 1) + hi;       // k-pair index
#pragma unroll
            for (int n = 0; n < 8; ++n) {
                const float2 bf = W1L[p * H_C + n * 16 + row];   // ds_load_b64
                v2f b; b.x = bf.x; b.y = bf.y;
                acc[n] = __builtin_amdgcn_wmma_f32_16x16x4_f32(
                    false, a, false, b, (short)0, acc[n], false, false);
            }
        }
        // ReLU + stage h tile (row-major 16 x 128) in per-wave LDS slab
#pragma unroll
        for (int n = 0; n < 8; ++n) {
#pragma unroll
            for (int v = 0; v < 8; ++v) {
                float val = acc[n][v];
                val = val > 0.0f ? val : 0.0f;
                hbuf[(v + 8 * hi) * H_C + n * 16 + row] = val;
            }
        }

        // ---- Layer 2: out(16x64) = h(16x128) @ W2 + b2 ----
        v8f acc2[4];
#pragma unroll
        for (int n = 0; n < 4; ++n) {
            const float bv = bv2[n];
            acc2[n] = (v8f){bv, bv, bv, bv, bv, bv, bv, bv};
        }
#pragma unroll
        for (int k = 0; k < H_C; k += 4) {
            const float2 af = hbuf2[(row * H_C + k + 2 * hi) >> 1];  // ds_load_b64
            v2f a; a.x = af.x; a.y = af.y;
            const int p = (k >> 1) + hi;
#pragma unroll
            for (int n = 0; n < 4; ++n) {
                const float2 bf = W2L[p * D_C + n * 16 + row];       // ds_load_b64
                v2f b; b.x = bf.x; b.y = bf.y;
                acc2[n] = __builtin_amdgcn_wmma_f32_16x16x4_f32(
                    false, a, false, b, (short)0, acc2[n], false, false);
            }
        }
        // Store D fragments: VGPR v -> row (v + 8*hi), col (lane%16)
#pragma unroll
        for (int n = 0; n < 4; ++n) {
#pragma unroll
            for (int v = 0; v < 8; ++v) {
                out[(size_t)(m0 + v + 8 * hi) * D_C + n * 16 + row] = acc2[n][v];
            }
        }
    }
}

// ---------------------------------------------------------------------------
extern "C" void kernel_launch(void* const* d_in, const int* in_sizes, int n_in,
                              void* d_out, int out_size, void* d_ws, size_t ws_size,
                              hipStream_t stream) {
    const float*     x  = (const float*)d_in[0];
    const long long* ei = (const long long*)d_in[1];   // int64 edge_index [2, E]
    const float*     W1 = (const float*)d_in[2];
    const float*     b1 = (const float*)d_in[3];
    const float*     W2 = (const float*)d_in[4];
    const float*     b2 = (const float*)d_in[5];
    float*           out = (float*)d_out;

    const int n_nodes = in_sizes[0] / D_C;        // 100000
    const int n_edges = in_sizes[1] / 2;          // 1600000
    const size_t agg_bytes = (size_t)n_nodes * D_C * sizeof(float);

    // agg lives in workspace; fall back to in-place d_out (safe: each wave
    // fully loads its 16-row tile before any store to the same rows).
    float* agg = (ws_size >= agg_bytes) ? (float*)d_ws : out;

    // 1) agg = x
    const int n4 = n_nodes * D_C / 4;
    init_agg_kernel<<<(n4 + 255) / 256, 256, 0, stream>>>(
        (const float4*)x, (float4*)agg, n4);

    // 2) agg[dst] += x[src]  (one wave per edge, 4 waves/block)
    scatter_add_kernel<<<(n_edges + 3) / 4, 128, 0, stream>>>(
        x, ei, agg, n_edges);

    // 3) fused MLP (one wave per 16-row tile, 8 tiles per wave)
    const int n_tiles = (n_nodes + 15) / 16;      // 6250
    const int tiles_per_block = WAVES_PB * TILES_PW;
    gin_mlp_kernel<<<(n_tiles + tiles_per_block - 1) / tiles_per_block, 128, 0, stream>>>(
        agg, W1, b1, W2, b2, out, n_tiles);
}